// Sparse4DHead2nd_70248485094104
// MI455X (gfx1250) — compile-verified
//
#include <hip/hip_runtime.h>
#include <hip/hip_bf16.h>
#include <math.h>

typedef __attribute__((ext_vector_type(2))) float v2f;
typedef __attribute__((ext_vector_type(8))) float v8f;

#define NUM_CAMS   6
#define NUM_LEVELS 4
#define NUM_PTS    13
#define NUM_GROUPS 8
#define NUM_ANCHOR 900
#define EMBED_DIMS 256
#define K_OUT      (NUM_CAMS * NUM_LEVELS * NUM_PTS * NUM_GROUPS)   // 2496
#define NUM_SLOTS  (NUM_CAMS * NUM_LEVELS * NUM_PTS)                // 312
#define TOTAL_HW   14960

// Level tables (H, W, flat start offset into feat's HW axis)
__constant__ int c_H[NUM_LEVELS]     = {64, 32, 16, 8};
__constant__ int c_W[NUM_LEVELS]     = {176, 88, 44, 22};
__constant__ int c_start[NUM_LEVELS] = {0, 11264, 14080, 14784};

// ---------------------------------------------------------------------------
// 1) X = instance_feature + anchor_embed
// ---------------------------------------------------------------------------
__global__ void add_embed_kernel(const float* __restrict__ inst,
                                 const float* __restrict__ anch,
                                 float* __restrict__ X) {
    int i = blockIdx.x * blockDim.x + threadIdx.x;
    if (i < NUM_ANCHOR * EMBED_DIMS) X[i] = inst[i] + anch[i];
}

// ---------------------------------------------------------------------------
// 2) fp32 WMMA GEMM:  C[M x N] = A[M x K] * B[N x K]^T + bias[N]
//    One wave computes a 16x64 strip (4 N-tiles). Per K-step: load the A
//    fragment and ALL FOUR B fragments into distinct registers FIRST, then
//    issue 4 back-to-back V_WMMA_F32_16X16X4_F32 — lets the compiler clause
//    the loads and stagger s_wait_loadcnt so WMMAs overlap memory latency.
//    A fragment (16x4): lanes 0-15 = rows, VGPR0/1 = K+0/K+1;
//                       lanes 16-31 = rows, VGPR0/1 = K+2/K+3.
//    B fragment (4x16): mirrored (lane%16 = column, same K split).
//    C/D (16x16 f32):   VGPR j -> row j (lanes 0-15) / row j+8 (lanes 16-31).
// ---------------------------------------------------------------------------
__global__ void gemm_wmma_f32(const float* __restrict__ A,
                              const float* __restrict__ B,
                              const float* __restrict__ bias,
                              float* __restrict__ C,
                              int M, int N, int K) {
    const int lane = threadIdx.x & 31;
    const int m0 = blockIdx.x * 16;
    const int n0 = blockIdx.y * 64;          // 4 tiles of 16

    const int halflane = lane & 15;
    const int khalf    = (lane >> 4) * 2;    // 0 or 2

    int arow = m0 + halflane;
    if (arow >= M) arow = M - 1;             // clamp pad rows (never stored)

    const float* __restrict__ aptr  = A + (size_t)arow * K + khalf;
    const float* __restrict__ bptr0 = B + (size_t)(n0 +  0 + halflane) * K + khalf;
    const float* __restrict__ bptr1 = B + (size_t)(n0 + 16 + halflane) * K + khalf;
    const float* __restrict__ bptr2 = B + (size_t)(n0 + 32 + halflane) * K + khalf;
    const float* __restrict__ bptr3 = B + (size_t)(n0 + 48 + halflane) * K + khalf;

    v8f acc0 = {}, acc1 = {}, acc2 = {}, acc3 = {};
    for (int k = 0; k < K; k += 4) {
        v2f af, b0, b1, b2, b3;
        // Issue all loads before any WMMA (distinct dest registers).
        af.x = aptr[k];      af.y = aptr[k + 1];
        b0.x = bptr0[k];     b0.y = bptr0[k + 1];
        b1.x = bptr1[k];     b1.y = bptr1[k + 1];
        b2.x = bptr2[k];     b2.y = bptr2[k + 1];
        b3.x = bptr3[k];     b3.y = bptr3[k + 1];
        acc0 = __builtin_amdgcn_wmma_f32_16x16x4_f32(false, af, false, b0,
                                                     (short)0, acc0, false, false);
        acc1 = __builtin_amdgcn_wmma_f32_16x16x4_f32(false, af, false, b1,
                                                     (short)0, acc1, false, false);
        acc2 = __builtin_amdgcn_wmma_f32_16x16x4_f32(false, af, false, b2,
                                                     (short)0, acc2, false, false);
        acc3 = __builtin_amdgcn_wmma_f32_16x16x4_f32(false, af, false, b3,
                                                     (short)0, acc3, false, false);
    }

    const int mbase = m0 + ((lane >> 4) ? 8 : 0);
    const int nn    = n0 + halflane;
#pragma unroll
    for (int j = 0; j < 8; ++j) {
        const int m = mbase + j;
        if (m < M) {
            float* crow = C + (size_t)m * N + nn;
            crow[0]  = acc0[j] + bias[nn];
            crow[16] = acc1[j] + bias[nn + 16];
            crow[32] = acc2[j] + bias[nn + 32];
            crow[48] = acc3[j] + bias[nn + 48];
        }
    }
}

// ---------------------------------------------------------------------------
// 3) Softmax over the 312 (cam,level,pt) slots per (anchor, group), in place.
//    One wave32 per row; 8 waves per block.
// ---------------------------------------------------------------------------
__global__ void softmax_kernel(float* __restrict__ logits) {
    const int row  = blockIdx.x * 8 + (threadIdx.x >> 5);   // 0 .. 7199
    const int lane = threadIdx.x & 31;
    if (row >= NUM_ANCHOR * NUM_GROUPS) return;
    const int a = row >> 3;
    const int g = row & 7;
    float* base = logits + (size_t)a * K_OUT + g;

    float m = -INFINITY;
    for (int i = lane; i < NUM_SLOTS; i += 32)
        m = fmaxf(m, base[i * NUM_GROUPS]);
#pragma unroll
    for (int off = 16; off > 0; off >>= 1)
        m = fmaxf(m, __shfl_xor(m, off, 32));

    float s = 0.f;
    for (int i = lane; i < NUM_SLOTS; i += 32)
        s += __expf(base[i * NUM_GROUPS] - m);
#pragma unroll
    for (int off = 16; off > 0; off >>= 1)
        s += __shfl_xor(s, off, 32);

    const float inv = 1.f / s;
    for (int i = lane; i < NUM_SLOTS; i += 32)
        base[i * NUM_GROUPS] = __expf(base[i * NUM_GROUPS] - m) * inv;
}

// ---------------------------------------------------------------------------
// 4) Project key points through lidar2img -> normalized 2D locations.
//    loc layout: [cam][anchor][pt][2]
// ---------------------------------------------------------------------------
__global__ void project_kernel(const float* __restrict__ key_points,
                               const float* __restrict__ lidar2img,
                               const float* __restrict__ image_wh,
                               float* __restrict__ loc) {
    const int idx = blockIdx.x * blockDim.x + threadIdx.x;
    if (idx >= NUM_CAMS * NUM_ANCHOR * NUM_PTS) return;
    const int cam = idx / (NUM_ANCHOR * NUM_PTS);
    const int rem = idx - cam * (NUM_ANCHOR * NUM_PTS);
    const int a   = rem / NUM_PTS;
    const int p   = rem - a * NUM_PTS;

    const float* pt = key_points + ((size_t)a * NUM_PTS + p) * 3;
    const float* Mx = lidar2img + (size_t)cam * 16;
    const float x = pt[0], y = pt[1], z = pt[2];
    const float px = Mx[0] * x + Mx[1] * y + Mx[2]  * z + Mx[3];
    const float py = Mx[4] * x + Mx[5] * y + Mx[6]  * z + Mx[7];
    float       pz = Mx[8] * x + Mx[9] * y + Mx[10] * z + Mx[11];
    pz = fmaxf(pz, 1e-5f);   // jnp.clip(z, 1e-5, None)

    loc[(size_t)idx * 2 + 0] = (px / pz) / image_wh[cam * 2 + 0];
    loc[(size_t)idx * 2 + 1] = (py / pz) / image_wh[cam * 2 + 1];
}

// ---------------------------------------------------------------------------
// 5) Fused bilinear gather + softmax-weighted accumulation.
//    Block = one anchor, thread = one channel. Tap indices/weights for all
//    312 (cam,pt,level) slots are staged ONCE in LDS (~10 KB), eliminating
//    the 256x-redundant per-thread bilinear math. Invalid taps get weight 0
//    with clamped indices (== reference's clip+valid-mask). Each wave maps
//    to exactly one group (c>>5), so the softmax weight is wave-uniform.
// ---------------------------------------------------------------------------
__global__ void sample_accumulate_kernel(const float* __restrict__ feat,
                                         const float* __restrict__ loc,
                                         const float* __restrict__ wts,
                                         float* __restrict__ features) {
    __shared__ int   s_idx[NUM_SLOTS][4];   // feat row index (within cam)
    __shared__ float s_tw[NUM_SLOTS][4];    // tap weight * validity

    const int a   = blockIdx.x;
    const int tid = threadIdx.x;            // 0..255

    // Stage per-slot tap data. Slot order: s = (cam*NUM_PTS + p)*NUM_LEVELS + l
    for (int s = tid; s < NUM_SLOTS; s += EMBED_DIMS) {
        const int l   = s & 3;
        const int cp  = s >> 2;
        const int p   = cp % NUM_PTS;
        const int cam = cp / NUM_PTS;

        const int lidx = (cam * NUM_ANCHOR + a) * NUM_PTS + p;
        const float u = loc[(size_t)lidx * 2 + 0];
        const float v = loc[(size_t)lidx * 2 + 1];

        const int H = c_H[l], W = c_W[l], st = c_start[l];
        const float xf = u * (float)W - 0.5f;
        const float yf = v * (float)H - 0.5f;
        const float x0f = floorf(xf), y0f = floorf(yf);
        const float wx1 = xf - x0f,  wy1 = yf - y0f;
        const float wx0 = 1.f - wx1, wy0 = 1.f - wy1;
        const int x0 = (int)x0f, y0 = (int)y0f;

        const float tw[4] = {wx0 * wy0, wx1 * wy0, wx0 * wy1, wx1 * wy1};
        const int   xs[4] = {x0, x0 + 1, x0,     x0 + 1};
        const int   ys[4] = {y0, y0,     y0 + 1, y0 + 1};
#pragma unroll
        for (int t = 0; t < 4; ++t) {
            const int xi = xs[t], yi = ys[t];
            const bool ok = (xi >= 0) & (xi < W) & (yi >= 0) & (yi < H);
            const int xc = min(max(xi, 0), W - 1);
            const int yc = min(max(yi, 0), H - 1);
            s_idx[s][t] = st + yc * W + xc;
            s_tw[s][t]  = ok ? tw[t] : 0.f;
        }
    }
    __syncthreads();

    const int c = tid;
    const int g = c >> 5;                   // wave-uniform group
    float acc = 0.f;
    for (int cam = 0; cam < NUM_CAMS; ++cam) {
        const float* __restrict__ featc =
            feat + ((size_t)cam * TOTAL_HW) * EMBED_DIMS + c;
        for (int p = 0; p < NUM_PTS; ++p) {
#pragma unroll
            for (int l = 0; l < NUM_LEVELS; ++l) {
                const int s = (cam * NUM_PTS + p) * NUM_LEVELS + l;
                const float wgt =
                    wts[(((size_t)(a * NUM_CAMS + cam) * NUM_LEVELS + l)
                             * NUM_PTS + p) * NUM_GROUPS + g];
                float bil = 0.f;
#pragma unroll
                for (int t = 0; t < 4; ++t)
                    bil += s_tw[s][t] * featc[(size_t)s_idx[s][t] * EMBED_DIMS];
                acc += wgt * bil;
            }
        }
    }
    features[(size_t)a * EMBED_DIMS + c] = acc;
}

// ---------------------------------------------------------------------------
// Host launcher
// ---------------------------------------------------------------------------
extern "C" void kernel_launch(void* const* d_in, const int* in_sizes, int n_in,
                              void* d_out, int out_size, void* d_ws, size_t ws_size,
                              hipStream_t stream) {
    const float* feat       = (const float*)d_in[0];
    const float* key_points = (const float*)d_in[1];
    const float* lidar2img  = (const float*)d_in[2];
    const float* image_wh   = (const float*)d_in[3];
    const float* inst       = (const float*)d_in[4];
    const float* anch       = (const float*)d_in[5];
    const float* wfc_w      = (const float*)d_in[6];
    const float* wfc_b      = (const float*)d_in[7];
    const float* op_w       = (const float*)d_in[8];
    const float* op_b       = (const float*)d_in[9];
    float*       out        = (float*)d_out;

    // Workspace carve-up (floats): X | logits/weights | loc | features
    float* X        = (float*)d_ws;
    float* logits   = X + (size_t)NUM_ANCHOR * EMBED_DIMS;                 // 900*2496
    float* loc      = logits + (size_t)NUM_ANCHOR * K_OUT;                 // 6*900*13*2
    float* features = loc + (size_t)NUM_CAMS * NUM_ANCHOR * NUM_PTS * 2;   // 900*256

    // 1) X = instance + anchor
    {
        const int n = NUM_ANCHOR * EMBED_DIMS;
        add_embed_kernel<<<(n + 255) / 256, 256, 0, stream>>>(inst, anch, X);
    }
    // 2) logits = X * Wfc^T + b   (M=900, N=2496, K=256); 64-wide N strips
    {
        dim3 grid((NUM_ANCHOR + 15) / 16, K_OUT / 64);
        gemm_wmma_f32<<<grid, 32, 0, stream>>>(X, wfc_w, wfc_b, logits,
                                               NUM_ANCHOR, K_OUT, EMBED_DIMS);
    }
    // 3) softmax over the 312 slots per (anchor, group), in place
    softmax_kernel<<<NUM_ANCHOR, 256, 0, stream>>>(logits);
    // 4) project key points
    {
        const int n = NUM_CAMS * NUM_ANCHOR * NUM_PTS;
        project_kernel<<<(n + 255) / 256, 256, 0, stream>>>(key_points, lidar2img,
                                                            image_wh, loc);
    }
    // 5) fused bilinear sample + weighted accumulate (LDS-staged taps)
    sample_accumulate_kernel<<<NUM_ANCHOR, EMBED_DIMS, 0, stream>>>(feat, loc,
                                                                    logits, features);
    // 6) out = features * out_proj_w^T + b   (M=900, N=256, K=256)
    {
        dim3 grid((NUM_ANCHOR + 15) / 16, EMBED_DIMS / 64);
        gemm_wmma_f32<<<grid, 32, 0, stream>>>(features, op_w, op_b, out,
                                               NUM_ANCHOR, EMBED_DIMS, EMBED_DIMS);
    }
}